// Denoising_27092653703705
// MI455X (gfx1250) — compile-verified
//
#include <hip/hip_runtime.h>
#include <hip/hip_bf16.h>

typedef __attribute__((ext_vector_type(2))) float v2f;
typedef __attribute__((ext_vector_type(8))) float v8f;

#define IN_F   256
#define HID    128
#define OUT_F  64
#define ATTN_SLOPE 0.2f
#define ACT_SLOPE  0.01f

__device__ __forceinline__ float lrelu(float v, float s) { return v > 0.f ? v : s * v; }

// Monotone float atomic max: int compare for >=0, unsigned compare (reversed) for <0.
__device__ __forceinline__ void atomicMaxF(float* addr, float v) {
    if (v >= 0.f) atomicMax((int*)addr, __float_as_int(v));
    else          atomicMin((unsigned int*)addr, __float_as_uint(v));
}

// ---------------------------------------------------------------------------
// Kernel 1: feat = x @ W_gat   [N,256] x [256,128] -> [N,128], fp32 WMMA
// grid.x = ceil(N/16); block = 256 (8 waves, one 16-col tile of W each)
// ---------------------------------------------------------------------------
__global__ void feat_gemm_kernel(const float* __restrict__ x,
                                 const float* __restrict__ Wg,   // [256][128]
                                 float* __restrict__ feat, int N) {
    __shared__ float As[16 * 260];              // 16 rows x 256 cols, pad->260 (bank-conflict free)
    const int row0 = blockIdx.x * 16;
    const bool full = (row0 + 16 <= N);
    // cooperative load of the 16x256 A tile (coalesced)
    if (full) {
        for (int i = threadIdx.x; i < 16 * 256; i += 256) {
            int r = i >> 8, c = i & 255;
            As[r * 260 + c] = x[(size_t)(row0 + r) * IN_F + c];
        }
    } else {
        for (int i = threadIdx.x; i < 16 * 256; i += 256) {
            int r = i >> 8, c = i & 255;
            As[r * 260 + c] = (row0 + r < N) ? x[(size_t)(row0 + r) * IN_F + c] : 0.f;
        }
    }
    __syncthreads();

    const int lane = threadIdx.x & 31;
    const int wv   = threadIdx.x >> 5;
    const int n0   = wv * 16;                   // this wave's output column tile
    const int m16  = lane & 15;
    const int half = lane >> 4;

    v8f acc = {};
    #pragma unroll 4
    for (int k0 = 0; k0 < IN_F; k0 += 4) {
        const int kb = k0 + 2 * half;
        v2f a, b;
        a.x = As[m16 * 260 + kb];
        a.y = As[m16 * 260 + kb + 1];
        b.x = Wg[(size_t)kb       * HID + n0 + m16];
        b.y = Wg[(size_t)(kb + 1) * HID + n0 + m16];
        acc = __builtin_amdgcn_wmma_f32_16x16x4_f32(false, a, false, b,
                                                    (short)0, acc, false, false);
    }

    // D: VGPR r holds row (row0 + 8*half + r); consecutive rows are HID*4 B apart
    float* dp = feat + (size_t)(row0 + 8 * half) * HID + n0 + m16;
    if (full) {
        #pragma unroll
        for (int r = 0; r < 8; ++r) dp[(size_t)r * HID] = acc[r];
    } else {
        #pragma unroll
        for (int r = 0; r < 8; ++r)
            if (row0 + 8 * half + r < N) dp[(size_t)r * HID] = acc[r];
    }
}

// ---------------------------------------------------------------------------
// Kernel 2: el[n] = feat[n,:].attn_l ; er likewise. Wave-per-node, wave32 shfl.
// ---------------------------------------------------------------------------
__global__ void node_attn_kernel(const float* __restrict__ feat,
                                 const float* __restrict__ attn_l,
                                 const float* __restrict__ attn_r,
                                 float* __restrict__ el, float* __restrict__ er, int N) {
    const int lane = threadIdx.x & 31;
    const int n = blockIdx.x * 8 + (threadIdx.x >> 5);
    if (n >= N) return;
    const float4 f  = ((const float4*)(feat + (size_t)n * HID))[lane];
    const float4 al = ((const float4*)attn_l)[lane];
    const float4 ar = ((const float4*)attn_r)[lane];
    float sl = f.x * al.x + f.y * al.y + f.z * al.z + f.w * al.w;
    float sr = f.x * ar.x + f.y * ar.y + f.z * ar.z + f.w * ar.w;
    #pragma unroll
    for (int off = 16; off > 0; off >>= 1) {
        sl += __shfl_xor(sl, off, 32);
        sr += __shfl_xor(sr, off, 32);
    }
    if (lane == 0) { el[n] = sl; er[n] = sr; }
}

// ---------------------------------------------------------------------------
// Kernel 3: init agg<-bias, emax<- -inf, denom<-0 (must run every call)
// ---------------------------------------------------------------------------
__global__ void init_kernel(const float* __restrict__ bias,
                            float* __restrict__ agg,
                            float* __restrict__ emax,
                            float* __restrict__ denom, int N) {
    const int idx = blockIdx.x * blockDim.x + threadIdx.x;
    if (idx >= N * HID) return;
    const int h = idx & (HID - 1);
    agg[idx] = bias[h];
    if (h == 0) {
        const int n = idx >> 7;
        emax[n]  = -__builtin_huge_valf();
        denom[n] = 0.f;
    }
}

// ---------------------------------------------------------------------------
// Kernel 4: per-edge logits + segment max by dst
// ---------------------------------------------------------------------------
__global__ void edge_logit_kernel(const int* __restrict__ src, const int* __restrict__ dst,
                                  const float* __restrict__ el, const float* __restrict__ er,
                                  float* __restrict__ elog, float* __restrict__ emax, int E) {
    const int e = blockIdx.x * blockDim.x + threadIdx.x;
    if (e >= E) return;
    const int d = dst[e];
    float v = lrelu(el[src[e]] + er[d], ATTN_SLOPE);
    elog[e] = v;
    atomicMaxF(&emax[d], v);
}

// ---------------------------------------------------------------------------
// Kernel 5: alpha = exp(e - emax[dst]); denom[dst] += alpha (hw fp32 atomic)
// ---------------------------------------------------------------------------
__global__ void edge_alpha_kernel(const int* __restrict__ dst,
                                  const float* __restrict__ elog,
                                  const float* __restrict__ emax,
                                  float* __restrict__ alog,
                                  float* __restrict__ denom, int E) {
    const int e = blockIdx.x * blockDim.x + threadIdx.x;
    if (e >= E) return;
    const int d = dst[e];
    float a = __expf(elog[e] - emax[d]);
    alog[e] = a;
    unsafeAtomicAdd(&denom[d], a);          // global_atomic_add_f32, resolves in L2
}

// ---------------------------------------------------------------------------
// Kernel 6: wave-per-edge scatter: agg[dst] += (alpha/denom[dst]) * feat[src]
// 32 lanes x 4 features, coalesced 512B per edge; feat+agg are L2-resident.
// ---------------------------------------------------------------------------
__global__ void edge_scatter_kernel(const int* __restrict__ src, const int* __restrict__ dst,
                                    const float* __restrict__ alog,
                                    const float* __restrict__ denom,
                                    const float* __restrict__ feat,
                                    float* __restrict__ agg, int E) {
    const int e = blockIdx.x * 8 + (threadIdx.x >> 5);
    if (e >= E) return;
    const int lane = threadIdx.x & 31;
    const int s = src[e], d = dst[e];
    const float a = alog[e] / denom[d];
    const float* fs = feat + (size_t)s * HID;
    float* ag = agg + (size_t)d * HID;
    #pragma unroll
    for (int j = 0; j < 4; ++j) {
        const int h = lane + j * 32;
        unsafeAtomicAdd(&ag[h], a * fs[h]); // global_atomic_add_f32
    }
}

// ---------------------------------------------------------------------------
// Kernel 7: out = leakyrelu(agg, .01) @ W_lin^T   [N,128]x[128,64], fp32 WMMA
// grid.x = ceil(N/16); block = 128 (4 waves, one 16-col tile each)
// ---------------------------------------------------------------------------
__global__ void out_gemm_kernel(const float* __restrict__ agg,
                                const float* __restrict__ Wl,    // [64][128]
                                float* __restrict__ out, int N) {
    __shared__ float As[16 * 132];              // 16x128, pad->132 (bank-conflict free)
    const int row0 = blockIdx.x * 16;
    const bool full = (row0 + 16 <= N);
    if (full) {
        for (int i = threadIdx.x; i < 16 * 128; i += 128) {
            int r = i >> 7, c = i & 127;
            As[r * 132 + c] = lrelu(agg[(size_t)(row0 + r) * HID + c], ACT_SLOPE);
        }
    } else {
        for (int i = threadIdx.x; i < 16 * 128; i += 128) {
            int r = i >> 7, c = i & 127;
            float v = (row0 + r < N) ? agg[(size_t)(row0 + r) * HID + c] : 0.f;
            As[r * 132 + c] = lrelu(v, ACT_SLOPE);
        }
    }
    __syncthreads();

    const int lane = threadIdx.x & 31;
    const int wv   = threadIdx.x >> 5;
    const int n0   = wv * 16;
    const int m16  = lane & 15;
    const int half = lane >> 4;

    v8f acc = {};
    #pragma unroll 4
    for (int k0 = 0; k0 < HID; k0 += 4) {
        const int kb = k0 + 2 * half;
        v2f a, b;
        a.x = As[m16 * 132 + kb];
        a.y = As[m16 * 132 + kb + 1];
        // B[k][n] = W_lin[n][k]
        b.x = Wl[(size_t)(n0 + m16) * HID + kb];
        b.y = Wl[(size_t)(n0 + m16) * HID + kb + 1];
        acc = __builtin_amdgcn_wmma_f32_16x16x4_f32(false, a, false, b,
                                                    (short)0, acc, false, false);
    }

    float* dp = out + (size_t)(row0 + 8 * half) * OUT_F + n0 + m16;
    if (full) {
        #pragma unroll
        for (int r = 0; r < 8; ++r) dp[(size_t)r * OUT_F] = acc[r];
    } else {
        #pragma unroll
        for (int r = 0; r < 8; ++r)
            if (row0 + 8 * half + r < N) dp[(size_t)r * OUT_F] = acc[r];
    }
}

// ---------------------------------------------------------------------------
extern "C" void kernel_launch(void* const* d_in, const int* in_sizes, int n_in,
                              void* d_out, int out_size, void* d_ws, size_t ws_size,
                              hipStream_t stream) {
    const float* x    = (const float*)d_in[0];
    const int*   src  = (const int*)d_in[1];
    const int*   dst  = (const int*)d_in[2];
    const float* Wg   = (const float*)d_in[3];
    const float* al   = (const float*)d_in[4];
    const float* ar   = (const float*)d_in[5];
    const float* bias = (const float*)d_in[6];
    const float* Wl   = (const float*)d_in[7];
    float* out = (float*)d_out;

    const int N = in_sizes[0] / IN_F;       // 50000
    const int E = in_sizes[1];              // 800000

    // Workspace carve-up (all 16B aligned): ~58 MB total
    char* ws = (char*)d_ws;
    float* feat  = (float*)ws; ws += (size_t)N * HID * sizeof(float);
    float* agg   = (float*)ws; ws += (size_t)N * HID * sizeof(float);
    float* elv   = (float*)ws; ws += (size_t)N * sizeof(float);
    float* erv   = (float*)ws; ws += (size_t)N * sizeof(float);
    float* emax  = (float*)ws; ws += (size_t)N * sizeof(float);
    float* denom = (float*)ws; ws += (size_t)N * sizeof(float);
    float* elog  = (float*)ws; ws += (size_t)E * sizeof(float);
    float* alog  = (float*)ws; ws += (size_t)E * sizeof(float);

    const int rowTiles = (N + 15) / 16;

    feat_gemm_kernel<<<rowTiles, 256, 0, stream>>>(x, Wg, feat, N);
    node_attn_kernel<<<(N + 7) / 8, 256, 0, stream>>>(feat, al, ar, elv, erv, N);
    init_kernel<<<(N * HID + 255) / 256, 256, 0, stream>>>(bias, agg, emax, denom, N);
    edge_logit_kernel<<<(E + 255) / 256, 256, 0, stream>>>(src, dst, elv, erv, elog, emax, E);
    edge_alpha_kernel<<<(E + 255) / 256, 256, 0, stream>>>(dst, elog, emax, alog, denom, E);
    edge_scatter_kernel<<<(E + 7) / 8, 256, 0, stream>>>(src, dst, alog, denom, feat, agg, E);
    out_gemm_kernel<<<rowTiles, 128, 0, stream>>>(agg, Wl, out, N);
}